// PPIGAT_38714835206278
// MI455X (gfx1250) — compile-verified
//
#include <hip/hip_runtime.h>
#include <hip/hip_bf16.h>
#include <math.h>

typedef float v2f __attribute__((ext_vector_type(2)));
typedef float v8f __attribute__((ext_vector_type(8)));

// ---------------------------------------------------------------------------
// GEMM via V_WMMA_F32_16X16X4_F32, fully padded shapes -> ZERO bounds checks.
// C[M,N] = A[M,K] * B[K,N], with B supplied TRANSPOSED: Bt[N][K] (row-major).
// One wave computes a 64(M) x 64(N) tile: 16 accumulator tiles, and per k-step
// only 8 x global_load_b64 (4 A frags + 4 B frags) feed 16 WMMAs.
// Requirements (guaranteed by caller): M % 64 == 0, N % 64 == 0, K % 4 == 0,
// lda/ldbt even (b64-aligned fragment loads).
// Fragment layouts:
//   A (16x4): lane L holds A [m0+(L&15)] [k + (L>>4)*2 + {0,1}] -> one b64
//   B (4x16): lane L holds Bt[n0+(L&15)] [k + (L>>4)*2 + {0,1}] -> one b64
//   C (16x16): VGPR r: lanes0-15 -> M=r, lanes16-31 -> M=r+8; N = lane&15
// ---------------------------------------------------------------------------
__global__ __launch_bounds__(32) void gemm_wmma_f32(
    const float* __restrict__ A, const float* __restrict__ Bt, float* __restrict__ C,
    int K, int lda, int ldbt, int ldc)
{
  const int lane = threadIdx.x & 31;
  const int m0 = blockIdx.y << 6;
  const int n0 = blockIdx.x << 6;
  const int rc = lane & 15;          // A-row / B,C-col within tile
  const int kb = (lane >> 4) << 1;   // k sub-offset within 4-step: 0 or 2

  v8f acc[4][4] = {};

  const float* Ap[4];
  const float* Bp[4];
#pragma unroll
  for (int t = 0; t < 4; ++t)
    Ap[t] = A + (long)(m0 + (t << 4) + rc) * lda + kb;
#pragma unroll
  for (int j = 0; j < 4; ++j)
    Bp[j] = Bt + (long)(n0 + (j << 4) + rc) * ldbt + kb;

  for (int k = 0; k < K; k += 4) {
    v2f a[4], b[4];
#pragma unroll
    for (int t = 0; t < 4; ++t) a[t] = *(const v2f*)(Ap[t] + k);   // b64
#pragma unroll
    for (int j = 0; j < 4; ++j) b[j] = *(const v2f*)(Bp[j] + k);   // b64
#pragma unroll
    for (int t = 0; t < 4; ++t)
#pragma unroll
      for (int j = 0; j < 4; ++j)
        acc[t][j] = __builtin_amdgcn_wmma_f32_16x16x4_f32(
            false, a[t], false, b[j], (short)0, acc[t][j], false, false);
  }

  const int mbase = m0 + ((lane >> 4) << 3);
#pragma unroll
  for (int t = 0; t < 4; ++t) {
#pragma unroll
    for (int j = 0; j < 4; ++j) {
      float* Cp = C + (long)(mbase + (t << 4)) * ldc + n0 + (j << 4) + rc;
#pragma unroll
      for (int r = 0; r < 8; ++r)
        Cp[(long)r * ldc] = acc[t][j][r];
    }
  }
}

// ---------------------------------------------------------------------------
// Zero-padded 2D copy: dst[dstR x dstC] <- src[srcR x srcC], zeros elsewhere.
// ---------------------------------------------------------------------------
__global__ void pad2d_kernel(const float* __restrict__ src, float* __restrict__ dst,
                             int srcR, int srcC, int dstR, int dstC)
{
  long i = (long)blockIdx.x * blockDim.x + threadIdx.x;
  const long total = (long)dstR * dstC;
  if (i >= total) return;
  const int r = (int)(i / dstC);
  const int c = (int)(i - (long)r * dstC);
  dst[i] = (r < srcR && c < srcC) ? src[(long)r * srcC + c] : 0.0f;
}

// Zero-padded transpose: dst[dstR x dstC] <- src^T, src is [srcR x srcC].
// dst[r][c] = src[c][r] when in range, else 0.  (dst row r = src column r)
__global__ void padT_kernel(const float* __restrict__ src, float* __restrict__ dst,
                            int srcR, int srcC, int dstR, int dstC)
{
  long i = (long)blockIdx.x * blockDim.x + threadIdx.x;
  const long total = (long)dstR * dstC;
  if (i >= total) return;
  const int r = (int)(i / dstC);
  const int c = (int)(i - (long)r * dstC);
  dst[i] = (c < srcR && r < srcC) ? src[(long)c * srcC + r] : 0.0f;
}

// ---------------------------------------------------------------------------
// CSR build over destination node (rebuilt each call -> no cross-call state).
// ---------------------------------------------------------------------------
__global__ void count_kernel(const int* __restrict__ eidx, int E, int* counts)
{
  int e = blockIdx.x * blockDim.x + threadIdx.x;
  if (e < E) atomicAdd(&counts[eidx[E + e]], 1);  // dst = eidx[E + e]
}

__global__ void scan_kernel(const int* __restrict__ counts, int* __restrict__ rowptr,
                            int* __restrict__ rowfill, int n)
{
  __shared__ int sh[1024];
  const int t = threadIdx.x;
  const int chunk = (n + 1023) / 1024;
  const int lo = t * chunk;
  const int hi = (lo + chunk < n) ? (lo + chunk) : n;
  int sum = 0;
  for (int i = lo; i < hi; ++i) sum += counts[i];
  sh[t] = sum;
  __syncthreads();
  for (int off = 1; off < 1024; off <<= 1) {
    int v = (t >= off) ? sh[t - off] : 0;
    __syncthreads();
    sh[t] += v;
    __syncthreads();
  }
  int run = sh[t] - sum;  // exclusive prefix
  for (int i = lo; i < hi; ++i) {
    rowptr[i] = run;
    rowfill[i] = run;
    run += counts[i];
  }
  if (t == 1023) rowptr[n] = sh[1023];
}

__global__ void scatter_kernel(const int* __restrict__ eidx, int E,
                               int* __restrict__ rowfill, int* __restrict__ esort)
{
  int e = blockIdx.x * blockDim.x + threadIdx.x;
  if (e < E) {
    int d = eidx[E + e];
    int pos = atomicAdd(&rowfill[d], 1);
    esort[pos] = e;
  }
}

// ---------------------------------------------------------------------------
// Per-(node,head) attention dot products: one wave per (n,h), butterfly reduce.
// att_s[n,h] = sum_f Wh[n*ldw + h*F + f] * a[h,f];  att_d uses a[h, F+f].
// ---------------------------------------------------------------------------
__global__ void attn_kernel(const float* __restrict__ Wh, int ldw,
                            const float* __restrict__ a,
                            float* __restrict__ att_s, float* __restrict__ att_d,
                            int N, int H, int F)
{
  const int wid = (int)((blockIdx.x * blockDim.x + threadIdx.x) >> 5);
  const int lane = threadIdx.x & 31;
  if (wid >= N * H) return;
  const int n = wid / H;
  const int h = wid - n * H;
  const float* row = Wh + (long)n * ldw + (long)h * F;
  const float* as = a + (long)h * 2 * F;
  const float* ad = as + F;
  float s = 0.0f, d = 0.0f;
  for (int f = lane; f < F; f += 32) {
    float v = row[f];
    s += v * as[f];
    d += v * ad[f];
  }
  for (int off = 16; off; off >>= 1) {
    s += __shfl_xor(s, off, 32);
    d += __shfl_xor(d, off, 32);
  }
  if (lane == 0) { att_s[wid] = s; att_d[wid] = d; }
}

// ---------------------------------------------------------------------------
// Numerically stable softmax over each destination's incoming edges.
// One thread per (node, head); avg degree is 16 so the loop is short.
// ---------------------------------------------------------------------------
__global__ void softmax_kernel(const int* __restrict__ esort, const int* __restrict__ rowptr,
                               const int* __restrict__ eidx, const float* __restrict__ att_s,
                               const float* __restrict__ att_d, float* __restrict__ coef,
                               int N, int H)
{
  const int t = blockIdx.x * blockDim.x + threadIdx.x;
  if (t >= N * H) return;
  const int n = t / H;
  const int h = t - n * H;
  const int start = rowptr[n], end = rowptr[n + 1];
  if (start == end) return;
  const float ad = att_d[n * H + h];
  float m = -3.402823466e38f;
  for (int i = start; i < end; ++i) {
    int e = esort[i];
    float l = att_s[eidx[e] * H + h] + ad;
    l = (l > 0.0f) ? l : 0.2f * l;               // leaky_relu, alpha=0.2
    m = fmaxf(m, l);
  }
  float denom = 0.0f;
  for (int i = start; i < end; ++i) {
    int e = esort[i];
    float l = att_s[eidx[e] * H + h] + ad;
    l = (l > 0.0f) ? l : 0.2f * l;
    float ex = expf(l - m);
    denom += ex;
    coef[(long)e * H + h] = ex;
  }
  const float inv = 1.0f / (denom + 1e-16f);
  for (int i = start; i < end; ++i) coef[(long)esort[i] * H + h] *= inv;
}

// ---------------------------------------------------------------------------
// Weighted aggregation, concat layout (H*F == 1024, F % 4 == 0):
//   out[n, c] = elu( sum_e coef[e, c/F] * Wh[src(e), c] ) (+ resid)
// One 256-thread block per node; each thread owns 4 CONSECUTIVE columns
// -> one b128 Wh load + one coef broadcast per edge per thread.
// ---------------------------------------------------------------------------
__global__ void agg_concat_kernel(const float* __restrict__ Wh, int ldw,
                                  const float* __restrict__ coef,
                                  const int* __restrict__ esort, const int* __restrict__ rowptr,
                                  const int* __restrict__ eidx, const float* __restrict__ resid,
                                  float* __restrict__ out, int H, int F)
{
  const int n = blockIdx.x;
  const int c0 = threadIdx.x << 2;          // 4 consecutive columns
  const int h = c0 / F;                     // uniform per thread
  float4 acc = make_float4(0.0f, 0.0f, 0.0f, 0.0f);
  const int start = rowptr[n], end = rowptr[n + 1];
  for (int i = start; i < end; ++i) {
    const int e = esort[i];
    const int s = eidx[e];                  // src node
    const float cf = coef[(long)e * H + h];
    const float4 wv = *(const float4*)(Wh + (long)s * ldw + c0);
    acc.x += cf * wv.x;
    acc.y += cf * wv.y;
    acc.z += cf * wv.z;
    acc.w += cf * wv.w;
  }
  // ELU (alpha = 1)
  acc.x = (acc.x > 0.0f) ? acc.x : (expf(acc.x) - 1.0f);
  acc.y = (acc.y > 0.0f) ? acc.y : (expf(acc.y) - 1.0f);
  acc.z = (acc.z > 0.0f) ? acc.z : (expf(acc.z) - 1.0f);
  acc.w = (acc.w > 0.0f) ? acc.w : (expf(acc.w) - 1.0f);
  const long ob = (long)n * (H * F) + c0;
  if (resid) {
    const float4 rv = *(const float4*)(resid + ob);
    acc.x += rv.x; acc.y += rv.y; acc.z += rv.z; acc.w += rv.w;
  }
  *(float4*)(out + ob) = acc;
}

// ---------------------------------------------------------------------------
// Final layer: aggregate then mean over heads. One 128-thread block per node.
// ---------------------------------------------------------------------------
__global__ void agg_mean_kernel(const float* __restrict__ Wh, int ldw,
                                const float* __restrict__ coef,
                                const int* __restrict__ esort, const int* __restrict__ rowptr,
                                const int* __restrict__ eidx, float* __restrict__ out,
                                int H, int F)
{
  const int n = blockIdx.x;
  const int f = threadIdx.x;
  if (f >= F) return;
  float acc = 0.0f;
  const int start = rowptr[n], end = rowptr[n + 1];
  for (int i = start; i < end; ++i) {
    const int e = esort[i];
    const int s = eidx[e];
    const float* wrow = Wh + (long)s * ldw;
#pragma unroll
    for (int h = 0; h < 6; ++h)
      acc += coef[(long)e * H + h] * wrow[h * F + f];
  }
  out[(long)n * F + f] = acc * (1.0f / (float)H);
}

// ---------------------------------------------------------------------------
extern "C" void kernel_launch(void* const* d_in, const int* in_sizes, int n_in,
                              void* d_out, int out_size, void* d_ws, size_t ws_size,
                              hipStream_t stream)
{
  const float* x  = (const float*)d_in[0];
  const int*   ei = (const int*)d_in[1];   // [2, E]: src = ei[0..E), dst = ei[E..2E)
  const float* W1 = (const float*)d_in[2];
  const float* a1 = (const float*)d_in[3];
  const float* W2 = (const float*)d_in[4];
  const float* a2 = (const float*)d_in[5];
  const float* W3 = (const float*)d_in[6];
  const float* a3 = (const float*)d_in[7];
  const int N  = in_sizes[0] / 50;         // 10000
  const int E  = in_sizes[1] / 2;          // 160000
  const int NP = (N + 63) & ~63;           // 10048 : M padded to x64
  const int KP1 = 52;                      // K=50 padded to x4
  const int NP3 = 768;                     // N=726 padded to x64
  float* out = (float*)d_out;

  // Workspace carve-out (~140 MB), 256B aligned slices.
  char* w = (char*)d_ws;
  auto take = [&](size_t bytes) -> char* {
    char* p = w;
    w += (bytes + 255) & ~(size_t)255;
    return p;
  };
  float* Wh    = (float*)take((size_t)NP * 1024 * sizeof(float)); // also holds NP x 768
  float* h1    = (float*)take((size_t)NP * 1024 * sizeof(float));
  float* h3    = (float*)take((size_t)NP * 1024 * sizeof(float));
  float* xpad  = (float*)take((size_t)NP * KP1 * sizeof(float));
  float* Bt1   = (float*)take((size_t)1024 * KP1 * sizeof(float)); // W1^T padded [1024 x 52]
  float* Bt2   = (float*)take((size_t)1024 * 1024 * sizeof(float));// W2^T        [1024 x 1024]
  float* Bt3   = (float*)take((size_t)NP3 * 1024 * sizeof(float)); // W3^T padded [768 x 1024]
  float* atts  = (float*)take((size_t)N * 6 * sizeof(float));
  float* attd  = (float*)take((size_t)N * 6 * sizeof(float));
  float* coef  = (float*)take((size_t)E * 6 * sizeof(float));
  int* rowptr  = (int*)take((size_t)(N + 1) * sizeof(int));
  int* rowfill = (int*)take((size_t)N * sizeof(int));
  int* counts  = (int*)take((size_t)N * sizeof(int));
  int* esort   = (int*)take((size_t)E * sizeof(int));

  // ---- Padded / transposed operand prep + zero pad rows of h1/h3 ----
  {
    long t1 = (long)NP * KP1;
    pad2d_kernel<<<(t1 + 255) / 256, 256, 0, stream>>>(x, xpad, N, 50, NP, KP1);
    long t2 = (long)1024 * KP1;  // Bt1[n][k] = W1[k][n], W1 is [50 x 1024]
    padT_kernel<<<(t2 + 255) / 256, 256, 0, stream>>>(W1, Bt1, 50, 1024, 1024, KP1);
    long t3 = (long)1024 * 1024; // Bt2[n][k] = W2[k][n], W2 is [1024 x 1024]
    padT_kernel<<<(t3 + 255) / 256, 256, 0, stream>>>(W2, Bt2, 1024, 1024, 1024, 1024);
    long t4 = (long)NP3 * 1024;  // Bt3[n][k] = W3[k][n], W3 is [1024 x 726]
    padT_kernel<<<(t4 + 255) / 256, 256, 0, stream>>>(W3, Bt3, 1024, 726, NP3, 1024);
    hipMemsetAsync(h1 + (size_t)N * 1024, 0, (size_t)(NP - N) * 1024 * sizeof(float), stream);
    hipMemsetAsync(h3 + (size_t)N * 1024, 0, (size_t)(NP - N) * 1024 * sizeof(float), stream);
  }

  // ---- CSR over destinations ----
  hipMemsetAsync(counts, 0, (size_t)N * sizeof(int), stream);
  count_kernel<<<(E + 255) / 256, 256, 0, stream>>>(ei, E, counts);
  scan_kernel<<<1, 1024, 0, stream>>>(counts, rowptr, rowfill, N);
  scatter_kernel<<<(E + 255) / 256, 256, 0, stream>>>(ei, E, rowfill, esort);

  // ---- Layer 1: [NP,52] x [52,1024]; 4 heads x 256, concat + ELU ----
  gemm_wmma_f32<<<dim3(1024 / 64, NP / 64), 32, 0, stream>>>(xpad, Bt1, Wh, KP1, KP1, KP1, 1024);
  attn_kernel<<<(N * 4 + 3) / 4, 128, 0, stream>>>(Wh, 1024, a1, atts, attd, N, 4, 256);
  softmax_kernel<<<(N * 4 + 255) / 256, 256, 0, stream>>>(esort, rowptr, ei, atts, attd, coef, N, 4);
  agg_concat_kernel<<<N, 256, 0, stream>>>(Wh, 1024, coef, esort, rowptr, ei, nullptr, h1, 4, 256);

  // ---- Layer 2: [NP,1024] x [1024,1024]; h3 = h1 + elu(gat) ----
  gemm_wmma_f32<<<dim3(1024 / 64, NP / 64), 32, 0, stream>>>(h1, Bt2, Wh, 1024, 1024, 1024, 1024);
  attn_kernel<<<(N * 4 + 3) / 4, 128, 0, stream>>>(Wh, 1024, a2, atts, attd, N, 4, 256);
  softmax_kernel<<<(N * 4 + 255) / 256, 256, 0, stream>>>(esort, rowptr, ei, atts, attd, coef, N, 4);
  agg_concat_kernel<<<N, 256, 0, stream>>>(Wh, 1024, coef, esort, rowptr, ei, h1, h3, 4, 256);

  // ---- Layer 3: [NP,1024] x [1024,768(726)]; 6 heads x 121, mean ----
  gemm_wmma_f32<<<dim3(NP3 / 64, NP / 64), 32, 0, stream>>>(h3, Bt3, Wh, 1024, 1024, 1024, NP3);
  attn_kernel<<<(N * 6 + 3) / 4, 128, 0, stream>>>(Wh, NP3, a3, atts, attd, N, 6, 121);
  softmax_kernel<<<(N * 6 + 255) / 256, 256, 0, stream>>>(esort, rowptr, ei, atts, attd, coef, N, 6);
  agg_mean_kernel<<<N, 128, 0, stream>>>(Wh, NP3, coef, esort, rowptr, ei, out, 6, 121);
}